// FastGTNs_60318520705505
// MI455X (gfx1250) — compile-verified
//
#include <hip/hip_runtime.h>
#include <hip/hip_bf16.h>

typedef __attribute__((ext_vector_type(2))) float v2f;
typedef __attribute__((ext_vector_type(8))) float v8f;

#define N_NODES   100000
#define T_TYPES   4
#define E_EDGES   400000
#define C_CH      2
#define L_LAYERS  2
#define W_IN_DIM  128
#define D_DIM     64
#define NUM_CLASS 16
#define BETA_V    0.5f
#define M_TGT     10000

// ---------------------------------------------------------------------------
// Tiny softmax over edge-type logits: filt[l,c,:] = softmax(conv_w[l,c,:])
// ---------------------------------------------------------------------------
__global__ void softmax_filt(const float* __restrict__ w, float* __restrict__ filt) {
    int i = threadIdx.x;                       // one thread per (l,c)
    if (i < L_LAYERS * C_CH) {
        const float* wi = w + i * T_TYPES;
        float mx = wi[0];
        for (int t = 1; t < T_TYPES; ++t) mx = fmaxf(mx, wi[t]);
        float e[T_TYPES]; float s = 0.f;
        for (int t = 0; t < T_TYPES; ++t) { e[t] = expf(wi[t] - mx); s += e[t]; }
        float inv = 1.0f / s;
        for (int t = 0; t < T_TYPES; ++t) filt[i * T_TYPES + t] = e[t] * inv;
    }
}

// ---------------------------------------------------------------------------
// fp32 WMMA GEMM: C[M,Ncol] = A[M,K] @ B[K,Ncol] (+bias), row-major, lda==K.
// 256 threads = 8 waves; each wave owns one 16x16 tile via v_wmma_f32_16x16x4_f32.
// Ncol must be a multiple of 16 and divide into <=8 col tiles; M multiple of 16.
// ---------------------------------------------------------------------------
__global__ __launch_bounds__(256) void gemm_wmma_f32(
    const float* __restrict__ A, const float* __restrict__ B,
    const float* __restrict__ bias, float* __restrict__ Cout,
    int M, int K, int Ncol)
{
    const int lane = threadIdx.x & 31;
    const int wave = threadIdx.x >> 5;
    const int colTiles = Ncol >> 4;
    const int rowTilesPerBlock = 8 / colTiles;
    const int rowTile = blockIdx.x * rowTilesPerBlock + wave / colTiles;
    const int colTile = wave - (wave / colTiles) * colTiles;
    if (rowTile * 16 >= M) return;             // wave-uniform: EXEC stays all-1s

    const int half = lane >> 4;                // 0: lanes 0-15, 1: lanes 16-31
    const int r    = lane & 15;

    const float* Arow = A + (size_t)(rowTile * 16 + r) * (size_t)K;
    const float* Bcol = B + (size_t)(colTile * 16 + r);

    v8f acc = {0.f, 0.f, 0.f, 0.f, 0.f, 0.f, 0.f, 0.f};
    for (int k0 = 0; k0 < K; k0 += 4) {
        const int ka = k0 + 2 * half;
        v2f a, b;
        a.x = Arow[ka];                        // A 16x4: VGPR0 K=k0(+2), VGPR1 K=+1(+3)
        a.y = Arow[ka + 1];
        b.x = Bcol[(size_t)ka * Ncol];         // B 4x16: lane = column
        b.y = Bcol[(size_t)(ka + 1) * Ncol];
        acc = __builtin_amdgcn_wmma_f32_16x16x4_f32(
                  false, a, false, b, (short)0, acc, false, false);
    }

    const float bv = bias ? bias[colTile * 16 + r] : 0.0f;
    float* Crow = Cout + (size_t)(rowTile * 16) * Ncol + colTile * 16;
#pragma unroll
    for (int v = 0; v < 8; ++v)                // C/D: VGPR v -> rows v and v+8
        Crow[(size_t)(v + 8 * half) * Ncol + r] = acc[v] + bv;
}

// ---------------------------------------------------------------------------
// Edge-parallel SpMM: Hout[c, row] += w(e,c) * Hin[c, col] (segment_sum dup-add)
// 16 threads per edge, each moves one float4 of the 64-float row.
// ---------------------------------------------------------------------------
__global__ __launch_bounds__(256) void spmm_edges(
    const float* __restrict__ Hin, float* __restrict__ Hout,
    const float* __restrict__ edge_value, const int* __restrict__ edge_index,
    const float* __restrict__ filt_l, long nUnits)
{
    long u = (long)blockIdx.x * blockDim.x + threadIdx.x;
    if (u >= nUnits) return;
    const int  c  = blockIdx.y;
    const long e  = u >> 4;                    // edge id in [0, T*E)
    const int  q  = (int)(u & 15);             // float4 slot in the D=64 row
    const int  t  = (int)(e / E_EDGES);
    const int  ep = (int)(e - (long)t * E_EDGES);
    const int  row = edge_index[((long)t * 2) * E_EDGES + ep];       // [t,0,ep]
    const int  col = edge_index[((long)t * 2 + 1) * E_EDGES + ep];   // [t,1,ep]
    const float w  = edge_value[e] * filt_l[c * T_TYPES + t];

    const float4 v = ((const float4*)(Hin + ((size_t)c * N_NODES + col) * D_DIM))[q];
    float* dst = Hout + ((size_t)c * N_NODES + row) * D_DIM + q * 4;
    unsafeAtomicAdd(dst + 0, w * v.x);         // global_atomic_add_f32
    unsafeAtomicAdd(dst + 1, w * v.y);
    unsafeAtomicAdd(dst + 2, w * v.z);
    unsafeAtomicAdd(dst + 3, w * v.w);
}

// ---------------------------------------------------------------------------
// Gather targets + residual + relu + channel-concat:
//   Hg[m, c*D+d] = relu(BETA*X_[c,tgt[m],d] + (1-BETA)*H[c,tgt[m],d])
// ---------------------------------------------------------------------------
__global__ __launch_bounds__(256) void gather_relu(
    const float* __restrict__ Xp, const float* __restrict__ H,
    const int* __restrict__ tgt, float* __restrict__ Hg)
{
    const long total = (long)M_TGT * C_CH * D_DIM;
    long i = (long)blockIdx.x * blockDim.x + threadIdx.x;
    if (i >= total) return;
    const int d = (int)(i % D_DIM);
    const int c = (int)((i / D_DIM) % C_CH);
    const int m = (int)(i / (C_CH * D_DIM));
    const int n = tgt[m];
    const size_t off = ((size_t)c * N_NODES + n) * D_DIM + d;
    const float v = BETA_V * Xp[off] + (1.0f - BETA_V) * H[off];
    Hg[i] = fmaxf(v, 0.0f);
}

// ---------------------------------------------------------------------------
extern "C" void kernel_launch(void* const* d_in, const int* in_sizes, int n_in,
                              void* d_out, int out_size, void* d_ws, size_t ws_size,
                              hipStream_t stream)
{
    const float* X          = (const float*)d_in[0];
    const float* edge_value = (const float*)d_in[1];
    const float* conv_w     = (const float*)d_in[2];
    const float* Ws         = (const float*)d_in[3];
    const float* lin1_w     = (const float*)d_in[4];
    const float* lin1_b     = (const float*)d_in[5];
    const float* lin_w      = (const float*)d_in[6];
    const float* lin_b      = (const float*)d_in[7];
    const int*   edge_index = (const int*)d_in[8];
    const int*   target_x   = (const int*)d_in[9];
    float* out = (float*)d_out;

    const size_t CND = (size_t)C_CH * N_NODES * D_DIM;   // 12.8M floats
    float* ws   = (float*)d_ws;
    float* filt = ws;              // 16 floats (padded to 64)
    float* Xp   = ws + 64;         // X_  [C,N,D]
    float* H1   = Xp + CND;        // layer-1 output
    float* H2   = H1 + CND;        // layer-2 output
    float* Hg   = H1;              // recycled: gathered target features [M, C*D]
    float* Htmp = H1 + (size_t)M_TGT * C_CH * D_DIM;     // [M, D]

    // 1) softmax over edge-type logits
    softmax_filt<<<1, 32, 0, stream>>>(conv_w, filt);

    // 2) X_[c] = X @ Ws[c]   (WMMA f32, N/16=6250 tiles, 2 row-tiles/block)
    for (int c = 0; c < C_CH; ++c)
        gemm_wmma_f32<<<N_NODES / 32, 256, 0, stream>>>(
            X, Ws + (size_t)c * W_IN_DIM * D_DIM, nullptr,
            Xp + (size_t)c * N_NODES * D_DIM, N_NODES, W_IN_DIM, D_DIM);

    // 3) L graph-propagation layers (edge-parallel gather + f32 atomic scatter)
    const long nUnits = (long)T_TYPES * E_EDGES * 16;    // 16 float4-units/edge
    const int  blocksE = (int)((nUnits + 255) / 256);
    const float* Hin = Xp;
    float* Hbuf[2] = {H1, H2};
    for (int l = 0; l < L_LAYERS; ++l) {
        float* Hout = Hbuf[l];
        hipMemsetAsync(Hout, 0, CND * sizeof(float), stream);
        spmm_edges<<<dim3(blocksE, C_CH), 256, 0, stream>>>(
            Hin, Hout, edge_value, edge_index,
            filt + l * C_CH * T_TYPES, nUnits);
        Hin = Hout;
    }

    // 4) gather targets + relu residual + concat channels -> Hg [M, 128]
    {
        const long nG = (long)M_TGT * C_CH * D_DIM;
        gather_relu<<<(int)((nG + 255) / 256), 256, 0, stream>>>(
            Xp, H2, target_x, Hg);
    }

    // 5) Htmp = Hg @ lin1_w + lin1_b   [10000,128]@[128,64]
    gemm_wmma_f32<<<(M_TGT / 16 + 1) / 2, 256, 0, stream>>>(
        Hg, lin1_w, lin1_b, Htmp, M_TGT, C_CH * D_DIM, D_DIM);

    // 6) out = Htmp @ lin_w + lin_b    [10000,64]@[64,16]
    gemm_wmma_f32<<<(M_TGT / 16 + 7) / 8, 256, 0, stream>>>(
        Htmp, lin_w, lin_b, out, M_TGT, D_DIM, NUM_CLASS);
}